// GaussianPointCloudContinuousPoseRasterisation_70970039599407
// MI455X (gfx1250) — compile-verified
//
#include <hip/hip_runtime.h>
#include <hip/hip_bf16.h>

#define NP 2048
#define HH 512
#define WW 512
#define CHUNK 256

typedef __attribute__((ext_vector_type(2))) float v2f;
typedef __attribute__((ext_vector_type(8))) float v8f;

// ---------------------------------------------------------------------------
// Kernel 1: quaternion -> rotation, camera transform, pinhole projection.
// Writes uvd workspace: [u_raw | v_raw | depth_raw], each NP floats.
// ---------------------------------------------------------------------------
__global__ void project_kernel(const float* __restrict__ pc,
                               const float* __restrict__ q,
                               const float* __restrict__ tc,
                               const float* __restrict__ K,
                               float* __restrict__ uvd) {
    int i = blockIdx.x * blockDim.x + threadIdx.x;
    if (i >= NP) return;
    float qx = q[0], qy = q[1], qz = q[2], qw = q[3];
    float r00 = 1.f - 2.f * (qy * qy + qz * qz);
    float r01 = 2.f * (qx * qy - qz * qw);
    float r02 = 2.f * (qx * qz + qy * qw);
    float r10 = 2.f * (qx * qy + qz * qw);
    float r11 = 1.f - 2.f * (qx * qx + qz * qz);
    float r12 = 2.f * (qy * qz - qx * qw);
    float r20 = 2.f * (qx * qz - qy * qw);
    float r21 = 2.f * (qy * qz + qx * qw);
    float r22 = 1.f - 2.f * (qx * qx + qy * qy);
    float px = pc[3 * i + 0], py = pc[3 * i + 1], pz = pc[3 * i + 2];
    float X = r00 * px + r01 * py + r02 * pz + tc[0];
    float Y = r10 * px + r11 * py + r12 * pz + tc[1];
    float Z = r20 * px + r21 * py + r22 * pz + tc[2];
    float fx = K[0], cx = K[2], fy = K[4], cy = K[5];
    float inv = 1.f / (Z + 1e-6f);
    uvd[i]            = (fx * X + cx * Z) * inv;
    uvd[NP + i]       = (fy * Y + cy * Z) * inv;
    uvd[2 * NP + i]   = Z;
}

// ---------------------------------------------------------------------------
// Kernel 2: single-block bitonic sort by depth (LDS) + gather sorted SoA.
// sorted layout (10 arrays of NP floats): u,v, ca,cb,cc, cr,cg,cb2, alpha, depth
// ---------------------------------------------------------------------------
__global__ __launch_bounds__(1024) void sort_gather_kernel(
        const float* __restrict__ uvd,
        const float* __restrict__ conic,
        const float* __restrict__ col,
        const float* __restrict__ alpha,
        float* __restrict__ sorted) {
    __shared__ float key[NP];
    __shared__ int   idx[NP];
    int t = threadIdx.x;
    for (int i = t; i < NP; i += 1024) { key[i] = uvd[2 * NP + i]; idx[i] = i; }
    __syncthreads();
    for (int k = 2; k <= NP; k <<= 1) {
        for (int j = k >> 1; j > 0; j >>= 1) {
            for (int i = t; i < NP; i += 1024) {
                int ixj = i ^ j;
                if (ixj > i) {
                    bool up = ((i & k) == 0);
                    float ki = key[i], kj = key[ixj];
                    bool sw = up ? (ki > kj) : (ki < kj);
                    if (sw) {
                        key[i] = kj; key[ixj] = ki;
                        int tmp = idx[i]; idx[i] = idx[ixj]; idx[ixj] = tmp;
                    }
                }
            }
            __syncthreads();
        }
    }
    for (int i = t; i < NP; i += 1024) {
        int s = idx[i];
        sorted[0 * NP + i] = uvd[s];
        sorted[1 * NP + i] = uvd[NP + s];
        sorted[2 * NP + i] = conic[3 * s + 0];
        sorted[3 * NP + i] = conic[3 * s + 1];
        sorted[4 * NP + i] = conic[3 * s + 2];
        sorted[5 * NP + i] = col[3 * s + 0];
        sorted[6 * NP + i] = col[3 * s + 1];
        sorted[7 * NP + i] = col[3 * s + 2];
        sorted[8 * NP + i] = alpha[s];
        sorted[9 * NP + i] = key[i];
    }
}

// ---------------------------------------------------------------------------
// Kernel 3: per-tile rasterization. One wave32 owns a 16x16 pixel tile.
// Per point: one V_WMMA_F32_16X16X4_F32 builds the full 16x16 power tile
//   power(m,n) = fy(m)*1 + 1*fx(n) + dy(m)*(-b*dx(n))
// (rank-3 contraction, K=4 with one unused slot), then per-lane exp +
// front-to-back compositing on the 8-VGPR C/D layout.
// ---------------------------------------------------------------------------
__global__ __launch_bounds__(128) void raster_kernel(const float* __restrict__ S,
                                                     float* __restrict__ out) {
    __shared__ float lds[10][CHUNK];
    const int tid   = threadIdx.x;
    const int wave  = tid >> 5;
    const int lane  = tid & 31;
    const int half  = lane >> 4;    // lane half selects K-pair {0,1} vs {2,3}
    const int idx16 = lane & 15;    // M (for A) and N (for B) index
    const int tx0 = blockIdx.x * 64 + wave * 16;
    const int ty0 = blockIdx.y * 16;
    const float pxn = (float)(tx0 + idx16) + 0.5f;  // pixel x for N=idx16
    const float pym = (float)(ty0 + idx16) + 0.5f;  // pixel y for M=idx16

    float T[8], Ra[8], Ga[8], Ba[8], Da[8];
#pragma unroll
    for (int v = 0; v < 8; ++v) { T[v] = 1.f; Ra[v] = Ga[v] = Ba[v] = Da[v] = 0.f; }

    for (int base = 0; base < NP; base += CHUNK) {
        __syncthreads();
        for (int j = tid; j < 10 * CHUNK; j += 128) {
            int arr = j / CHUNK, el = j % CHUNK;
            lds[arr][el] = S[arr * NP + base + el];
        }
        __syncthreads();

        for (int i = 0; i < CHUNK; ++i) {
            float u   = lds[0][i], vv = lds[1][i];
            float ca  = lds[2][i], cb_ = lds[3][i], cc = lds[4][i];
            float cr  = lds[5][i], cg = lds[6][i], cb2 = lds[7][i];
            float ai  = lds[8][i], di = lds[9][i];

            float dxn = pxn - u;
            float dym = pym - vv;
            float fyv = -0.5f * cc * dym * dym;
            float fxv = -0.5f * ca * dxn * dxn;

            // A(16x4): lanes 0-15 hold K={0,1} = {fy, 1}; lanes 16-31 K={2,3} = {dy, 0}
            // B(4x16): lanes 0-15 hold K={0,1} = {1, fx}; lanes 16-31 K={2,3} = {-b*dx, 0}
            // Branchless selects -> EXEC stays all-ones across the WMMA.
            v2f Av, Bv;
            Av.x = half ? dym : fyv;
            Av.y = half ? 0.f : 1.f;
            Bv.x = half ? (-cb_ * dxn) : 1.f;
            Bv.y = half ? 0.f : fxv;
            v8f cz = {};
            v8f pw = __builtin_amdgcn_wmma_f32_16x16x4_f32(
                false, Av, false, Bv, (short)0, cz, false, false);

#pragma unroll
            for (int v = 0; v < 8; ++v) {
                float p    = fminf(pw[v], 0.f);
                float g    = __builtin_amdgcn_exp2f(p * 1.4426950408889634f);
                float prod = g * ai;
                float a    = (prod >= (1.0f / 255.0f)) ? fminf(prod, 0.99f) : 0.f;
                float w    = T[v] * a;
                Ra[v] += w * cr;
                Ga[v] += w * cg;
                Ba[v] += w * cb2;
                Da[v] += w * di;
                T[v] *= (1.f - a);
            }
        }
    }

    // C/D layout: vgpr v of lane l -> pixel (M = v + 8*(l/16), N = l%16)
#pragma unroll
    for (int v = 0; v < 8; ++v) {
        int y = ty0 + v + 8 * half;
        int x = tx0 + idx16;
        float* o = out + ((size_t)(y * WW + x)) * 5;
        o[0] = Ra[v];
        o[1] = Ga[v];
        o[2] = Ba[v];
        o[3] = Da[v];
        o[4] = 1.f - T[v];
    }
}

extern "C" void kernel_launch(void* const* d_in, const int* in_sizes, int n_in,
                              void* d_out, int out_size, void* d_ws, size_t ws_size,
                              hipStream_t stream) {
    const float* pc    = (const float*)d_in[0];  // pointcloud      (NP,3)
    const float* col   = (const float*)d_in[1];  // point_color     (NP,3)
    const float* alpha = (const float*)d_in[2];  // point_alpha     (NP,)
    const float* conic = (const float*)d_in[3];  // point_conic     (NP,3)
    const float* q     = (const float*)d_in[4];  // quaternion      (4,)
    const float* tc    = (const float*)d_in[5];  // translation     (3,)
    const float* K     = (const float*)d_in[6];  // intrinsics      (3,3)
    (void)in_sizes; (void)n_in; (void)out_size; (void)ws_size;

    float* ws     = (float*)d_ws;
    float* uvd    = ws;            // 3*NP floats
    float* sorted = ws + 3 * NP;   // 10*NP floats

    project_kernel<<<(NP + 255) / 256, 256, 0, stream>>>(pc, q, tc, K, uvd);
    sort_gather_kernel<<<1, 1024, 0, stream>>>(uvd, conic, col, alpha, sorted);
    raster_kernel<<<dim3(WW / 64, HH / 16), 128, 0, stream>>>(sorted, (float*)d_out);
}